// FraudDetectionModel_65481071409764
// MI455X (gfx1250) — compile-verified
//
#include <hip/hip_runtime.h>
#include <math.h>

// MI455X / gfx1250, wave32.
//
// Reference collapses to: per-row scalar score s_i (tiny MLP), then
//   r_i = sum_{j != i} [ s_i*s_j / (|s_i||s_j| + 1e-12) >= 0.8 ] * s_j
//   out_i = r_i * dot(w_g, w_c) + b_c[0]
//
// HBM traffic ~192KB (~10ns at 23.3 TB/s): the real cost is the 16384^2
// pair predicates (VALU/matrix bound). Per 16x16 tile, two
// V_WMMA_F32_16X16X4_F32 rank-1 outer products:
//   num[m,n] = sj[m]*si[n]     (exact f32 single products)
//   sjb[m,n] = sj[m]*1.0       (sj broadcast in C-layout -> lane-local sum)
// Predicate algebra: den = |si||sj|+1e-12 == |num|+1e-12 bit-exactly
// (RNE rounding commutes with |.|), and num >= 0.8*(|num|+1e-12)
// <=> num >= 4e-12 (num<=0 -> false; num>0 -> 0.2*num >= 8e-13).
// So the adjacency test is ONE v_cmp_ge_f32 against a literal.
//
// Schedule: j-loop processes 2 tiles/iteration with the next pair of sj
// loads issued one iteration ahead (ds_load latency hidden behind the
// WMMA+VALU body instead of a per-tile s_wait_dscnt 0 stall).
// Occupancy: one block per i-tile (1024 blocks x 8 waves = 8192 waves);
// fixed-order LDS reduction combines the 8 j-chunk partials (deterministic).

typedef __attribute__((ext_vector_type(2))) float v2f;
typedef __attribute__((ext_vector_type(8))) float v8f;

#define ADJ_THRESH 4.0e-12f   // num >= 4e-12  <=>  fid >= 0.8

// ---------------- Kernel 1: per-row MLP score s_i --------------------------
__global__ __launch_bounds__(256)
void score_kernel(const float* __restrict__ x,
                  const float* __restrict__ w0, const float* __restrict__ b0,
                  const float* __restrict__ s0, const float* __restrict__ sh0,
                  const float* __restrict__ w1, const float* __restrict__ b1,
                  const float* __restrict__ s1, const float* __restrict__ sh1,
                  const float* __restrict__ w2, const float* __restrict__ b2,
                  const float* __restrict__ s2, const float* __restrict__ sh2,
                  const float* __restrict__ w_out, const float* __restrict__ b_out,
                  float* __restrict__ s, int n)
{
    int i = blockIdx.x * blockDim.x + threadIdx.x;
    if (i >= n) return;
    float h0 = x[2 * i], h1 = x[2 * i + 1];

    // layer: tanh(x @ w.T + b) * scale + shift ; w row-major (2,2)
    float t0 = fmaf(tanhf(fmaf(h1, w0[1], fmaf(h0, w0[0], b0[0]))), s0[0], sh0[0]);
    float t1 = fmaf(tanhf(fmaf(h1, w0[3], fmaf(h0, w0[2], b0[1]))), s0[1], sh0[1]);
    h0 = t0; h1 = t1;
    t0 = fmaf(tanhf(fmaf(h1, w1[1], fmaf(h0, w1[0], b1[0]))), s1[0], sh1[0]);
    t1 = fmaf(tanhf(fmaf(h1, w1[3], fmaf(h0, w1[2], b1[1]))), s1[1], sh1[1]);
    h0 = t0; h1 = t1;
    t0 = fmaf(tanhf(fmaf(h1, w2[1], fmaf(h0, w2[0], b2[0]))), s2[0], sh2[0]);
    t1 = fmaf(tanhf(fmaf(h1, w2[3], fmaf(h0, w2[2], b2[1]))), s2[1], sh2[1]);

    s[i] = fmaf(t1, w_out[1], fmaf(t0, w_out[0], b_out[0]));
}

// ---------------- Kernel 2: exact pairwise graph reduction -----------------
// One block per 16-wide i-tile; wave w scans j-chunk w (n/8 elements).
__global__ __launch_bounds__(256)
void graph_kernel(const float* __restrict__ s,
                  const float* __restrict__ w_g, const float* __restrict__ w_c,
                  const float* __restrict__ b_c,
                  float* __restrict__ out, int n)
{
    __shared__ float s_sh[16384 + 32];       // +32: prefetch overrun pad
    __shared__ float red[8 * 16];            // per-wave partials
    for (int k = threadIdx.x; k < n; k += blockDim.x) s_sh[k] = s[k];
    __syncthreads();

    const int  lane   = threadIdx.x & 31;
    const int  wave   = threadIdx.x >> 5;
    const int  i_base = blockIdx.x * 16;     // same i-tile for all 8 waves
    const bool lo     = lane < 16;
    const int  l15    = lane & 15;

    // B operands (fixed per block): 4x16, only row K=0 nonzero.
    // Assumed B layout mirrors the documented A 16x4 (VGPR0 lanes0-15 = K=0 row).
    const float si  = s_sh[i_base + l15];
    const v2f B_s   = { lo ? si   : 0.0f, 0.0f };
    const v2f B_one = { lo ? 1.0f : 0.0f, 0.0f };

    const int chunk = n >> 3;                // 2048 j's per wave
    const int j_lo  = wave * chunk;
    const int j_hi  = j_lo + chunk;

    // 4 accumulators -> compiler packs into v_pk_add_f32 pairs; fixed
    // element->accumulator mapping keeps the sum order deterministic.
    float a0 = 0.0f, a1 = 0.0f, a2 = 0.0f, a3 = 0.0f;

    // Software-pipelined: sjA/sjB for the current 2 tiles are loaded one
    // iteration ahead so ds_load latency overlaps the WMMA+VALU body.
    float sjA = s_sh[j_lo + l15];
    float sjB = s_sh[j_lo + 16 + l15];

    for (int j = j_lo; j < j_hi; j += 32) {
        const float pA = s_sh[j + 32 + l15];   // next-iteration prefetch
        const float pB = s_sh[j + 48 + l15];   // (pad keeps this in-bounds)

        v8f c0 = {};
        {   // tile A: A operand 16x4, column K=0 = sj (VGPR0 lanes0-15).
            v2f A = { lo ? sjA : 0.0f, 0.0f };
            v8f num = __builtin_amdgcn_wmma_f32_16x16x4_f32(false, A, false, B_s,   (short)0, c0, false, false);
            v8f sjb = __builtin_amdgcn_wmma_f32_16x16x4_f32(false, A, false, B_one, (short)0, c0, false, false);
            a0 += (num[0] >= ADJ_THRESH) ? sjb[0] : 0.0f;
            a1 += (num[1] >= ADJ_THRESH) ? sjb[1] : 0.0f;
            a2 += (num[2] >= ADJ_THRESH) ? sjb[2] : 0.0f;
            a3 += (num[3] >= ADJ_THRESH) ? sjb[3] : 0.0f;
            a0 += (num[4] >= ADJ_THRESH) ? sjb[4] : 0.0f;
            a1 += (num[5] >= ADJ_THRESH) ? sjb[5] : 0.0f;
            a2 += (num[6] >= ADJ_THRESH) ? sjb[6] : 0.0f;
            a3 += (num[7] >= ADJ_THRESH) ? sjb[7] : 0.0f;
        }
        {   // tile B
            v2f A = { lo ? sjB : 0.0f, 0.0f };
            v8f num = __builtin_amdgcn_wmma_f32_16x16x4_f32(false, A, false, B_s,   (short)0, c0, false, false);
            v8f sjb = __builtin_amdgcn_wmma_f32_16x16x4_f32(false, A, false, B_one, (short)0, c0, false, false);
            a0 += (num[0] >= ADJ_THRESH) ? sjb[0] : 0.0f;
            a1 += (num[1] >= ADJ_THRESH) ? sjb[1] : 0.0f;
            a2 += (num[2] >= ADJ_THRESH) ? sjb[2] : 0.0f;
            a3 += (num[3] >= ADJ_THRESH) ? sjb[3] : 0.0f;
            a0 += (num[4] >= ADJ_THRESH) ? sjb[4] : 0.0f;
            a1 += (num[5] >= ADJ_THRESH) ? sjb[5] : 0.0f;
            a2 += (num[6] >= ADJ_THRESH) ? sjb[6] : 0.0f;
            a3 += (num[7] >= ADJ_THRESH) ? sjb[7] : 0.0f;
        }
        sjA = pA;
        sjB = pB;
    }

    float acc = (a0 + a1) + (a2 + a3);

    // Lane n holds rows 0-7 of column n; lane n+16 holds rows 8-15. Combine.
    acc += __shfl_xor(acc, 16, 32);
    if (lo) red[wave * 16 + lane] = acc;
    __syncthreads();

    // Fixed-order cross-wave reduction + epilogue (deterministic).
    if (threadIdx.x < 16) {
        float r = 0.0f;
#pragma unroll
        for (int k = 0; k < 8; ++k) r += red[k * 16 + threadIdx.x];

        const int   i  = i_base + threadIdx.x;
        const float sv = s_sh[i];
        const float p  = sv * sv;                   // == in-loop diag num
        r -= (p >= ADJ_THRESH) ? sv : 0.0f;         // identical predicate ->
                                                    // exact diagonal cancel
        const float gc = w_g[0] * w_c[0] + w_g[1] * w_c[1]
                       + w_g[2] * w_c[2] + w_g[3] * w_c[3];
        out[i] = fmaf(r, gc, b_c[0]);
    }
}

// ---------------------------------------------------------------------------
extern "C" void kernel_launch(void* const* d_in, const int* in_sizes, int n_in,
                              void* d_out, int out_size, void* d_ws, size_t ws_size,
                              hipStream_t stream)
{
    const float* x     = (const float*)d_in[0];
    const float* w0    = (const float*)d_in[1];
    const float* b0    = (const float*)d_in[2];
    const float* s0    = (const float*)d_in[3];
    const float* sh0   = (const float*)d_in[4];
    const float* w1    = (const float*)d_in[5];
    const float* b1    = (const float*)d_in[6];
    const float* s1    = (const float*)d_in[7];
    const float* sh1   = (const float*)d_in[8];
    const float* w2    = (const float*)d_in[9];
    const float* b2    = (const float*)d_in[10];
    const float* s2    = (const float*)d_in[11];
    const float* sh2   = (const float*)d_in[12];
    const float* w_out = (const float*)d_in[13];
    const float* b_out = (const float*)d_in[14];
    const float* w_g   = (const float*)d_in[15];
    const float* w_c   = (const float*)d_in[16];
    const float* b_c   = (const float*)d_in[17];

    const int n = in_sizes[0] / 2;           // 16384
    float* s_ws = (float*)d_ws;              // N floats of scratch
    float* out  = (float*)d_out;

    score_kernel<<<(n + 255) / 256, 256, 0, stream>>>(
        x, w0, b0, s0, sh0, w1, b1, s1, sh1, w2, b2, s2, sh2,
        w_out, b_out, s_ws, n);

    // one block (8 waves) per 16-row i-tile; waves split the j-range 8-way
    graph_kernel<<<n / 16, 256, 0, stream>>>(s_ws, w_g, w_c, b_c, out, n);
}